// SpectralConv1d_83253646065780
// MI455X (gfx1250) — compile-verified
//
#include <hip/hip_runtime.h>
#include <math.h>
#include <stdint.h>

#define BB   16
#define CIN  128
#define COUT 128
#define LL   8192
#define MM   64
#define REG  1024
#define NR   8

typedef float v2f __attribute__((ext_vector_type(2)));
typedef float v8f __attribute__((ext_vector_type(8)));
typedef unsigned int u32x4 __attribute__((ext_vector_type(4)));
typedef int i32x4 __attribute__((ext_vector_type(4)));
typedef int i32x8 __attribute__((ext_vector_type(8)));

__device__ __forceinline__ float casv(float ang) {
  return cosf(ang) + sinf(ang);
}

// D = A(16x4 f32) * B(4x16 f32) + C, full fp32 WMMA
__device__ __forceinline__ v8f wmma4(v2f a, v2f b, v8f c) {
  return __builtin_amdgcn_wmma_f32_16x16x4_f32(
      /*neg_a=*/false, a, /*neg_b=*/false, b,
      /*c_mod=*/(short)0, c, /*reuse_a=*/false, /*reuse_b=*/false);
}

// TDM: DMA nelem4 dwords (<=65535) from global to LDS.
// Degenerate 2D descriptor: tile_dim0 = nelem4, tile_dim1 = 1.
__device__ __forceinline__ void tdm_load_to_lds(void* lds_dst, const void* gsrc,
                                                unsigned nelem4) {
  unsigned lds_addr = (unsigned)(uintptr_t)lds_dst;      // LDS offset in low 32b
  unsigned long long ga = (unsigned long long)(uintptr_t)gsrc;
  u32x4 g0;
  g0[0] = 1u;                                            // count=1, load, user mode
  g0[1] = lds_addr;                                      // D#.lds_addr
  g0[2] = (unsigned)(ga & 0xffffffffu);                  // global_addr[31:0]
  g0[3] = (unsigned)((ga >> 32) & 0x1ffffffu) | (2u << 30); // addr[56:32] | type=2
  i32x8 g1;
  g1[0] = (2 << 16);                                     // data_size = 4 bytes
  g1[1] = (int)((nelem4 & 0xffffu) << 16);               // tensor_dim0 low16
  g1[2] = (int)((nelem4 >> 16) & 0xffffu) | (1 << 16);   // dim0 hi16 | tensor_dim1=1
  g1[3] = (int)(nelem4 << 16);                           // tile_dim0 = nelem4
  g1[4] = 1;                                             // tile_dim1=1, tile_dim2=0
  g1[5] = (int)nelem4;                                   // tensor_dim0_stride low32
  g1[6] = (int)((nelem4 & 0xffffu) << 16);               // stride0 hi | stride1 lo
  g1[7] = 0;
  i32x4 gz4 = {0, 0, 0, 0};
  i32x8 gz8 = {0, 0, 0, 0, 0, 0, 0, 0};
  __builtin_amdgcn_tensor_load_to_lds(g0, g1, gz4, gz4, gz8, 0);
}

// ---------------- K0: cas tables + weight transpose ----------------
__global__ void prep_kernel(const float* __restrict__ w,
                            float* __restrict__ casML,   // [64][1024]
                            float* __restrict__ casLT,   // [1024][64]
                            float* __restrict__ Wbig) {  // [o][t*128+i]
  const float twopi = 6.28318530717958647692f;
  int gid = blockIdx.x * 256 + threadIdx.x;
  int stride = gridDim.x * 256;
  for (int idx = gid; idx < CIN * COUT * MM; idx += stride) {
    int o = idx >> 13;        // /8192
    int rem = idx & 8191;
    int t = rem >> 7;         // /128
    int i = rem & 127;
    Wbig[idx] = w[((size_t)i * COUT + o) * MM + t];
  }
  for (int idx = gid; idx < MM * REG; idx += stride) {
    int t = idx >> 10, l = idx & 1023;
    float ang = twopi * (float)(t * l) / (float)REG;
    float v = casv(ang);
    casML[t * REG + l] = v;
    casLT[l * MM + t]  = v;
  }
}

// ---------------- K1: per-(b,region) active mode count ----------------
__global__ void nm_kernel(const float* __restrict__ err, int* __restrict__ nmA) {
  __shared__ float part[256];
  __shared__ float avgs[BB];
  int r = blockIdx.x, tid = threadIdx.x;
  int b = tid >> 4, seg = tid & 15;
  float s = 0.f;
  const float* e = err + (size_t)b * LL + r * REG + seg * 64;
  for (int j = 0; j < 64; ++j) s += e[j];
  part[tid] = s;
  __syncthreads();
  if (tid < BB) {
    float t = 0.f;
    for (int j = 0; j < 16; ++j) t += part[tid * 16 + j];
    avgs[tid] = t / (float)REG;
  }
  __syncthreads();
  if (tid == 0) {
    float mn = avgs[0], mx = avgs[0];
    for (int j = 1; j < BB; ++j) { mn = fminf(mn, avgs[j]); mx = fmaxf(mx, avgs[j]); }
    float d = mx - mn;
    bool ok = d > 1e-8f;
    float denom = ok ? d : 1.f;
    for (int j = 0; j < BB; ++j) {
      float norm = ok ? (avgs[j] - mn) / denom : 0.f;
      int nm = (int)(norm * (float)(MM - 1)) + 1;
      nmA[j * NR + r] = nm;   // index by b*8 + r
    }
  }
}

// ---------------- K2: forward DHT (1024 -> 64 modes), WMMA GEMM ----------------
// X1[bid][c][t] = sum_l x[b][c][r*1024+l] * cas(2*pi*t*l/1024)
__global__ void __launch_bounds__(256)
dht_fwd_kernel(const float* __restrict__ x,
               const float* __restrict__ casLT,
               float* __restrict__ X1) {
  int bid = blockIdx.x;            // b*8 + r
  int b = bid >> 3, r = bid & 7;
  int tid = threadIdx.x;
  int wave = tid >> 5, lane = tid & 31;
  int n = lane & 15, hi = lane >> 4;
  int c0 = wave * 16;
  const float* xbase = x + (size_t)b * CIN * LL + (size_t)r * REG;
  v8f acc[4] = {};
  for (int ch = 0; ch < 256; ++ch) {
    int l = ch * 4;
    const float* ap = xbase + (size_t)(c0 + n) * LL + l + hi * 2;
    v2f a; a.x = ap[0]; a.y = ap[1];
#pragma unroll
    for (int kt = 0; kt < 4; ++kt) {
      int t0 = kt * 16;
      v2f bf;
      bf.x = casLT[(l + hi * 2) * MM + t0 + n];
      bf.y = casLT[(l + hi * 2 + 1) * MM + t0 + n];
      acc[kt] = wmma4(a, bf, acc[kt]);
    }
  }
#pragma unroll
  for (int kt = 0; kt < 4; ++kt)
#pragma unroll
    for (int v = 0; v < 8; ++v) {
      int c = c0 + v + hi * 8;
      int t = kt * 16 + n;
      X1[((size_t)bid * CIN + c) * MM + t] = acc[kt][v];
    }
}

// ---------------- K3: Z = Wbig(128 x 8192) @ Bt(8192 x 64), WMMA ----------------
__global__ void __launch_bounds__(256)
zmat_kernel(const float* __restrict__ X1,
            const float* __restrict__ Wbig,
            const int* __restrict__ nmA,
            float* __restrict__ Zm) {
  __shared__ float Xs[CIN * MM];     // 32KB: X1 then transformed X (in place)
  __shared__ float Bt[64 * MM];      // 16KB: half-i B tile
  __shared__ float cet[MM], cot_[MM], cvec[MM], maskf[MM];
  __shared__ int flip[MM];
  int bid = blockIdx.x;
  int tid = threadIdx.x;
  int nm = nmA[bid];
  float fnm = (float)nm;
  const float twopi = 6.28318530717958647692f;

  if (tid < MM) {
    int k = tid;
    maskf[k] = (k < nm) ? 1.f : 0.f;
    int f = (k == 0) ? 0 : (nm - k);
    f = f < 0 ? 0 : (f > 63 ? 63 : f);
    flip[k] = f;
  }
  // Tensor Data Mover: DMA the 128x64 f32 X1 panel straight into LDS.
  if (tid < 32) {
    tdm_load_to_lds(&Xs[0], X1 + (size_t)bid * CIN * MM, CIN * MM);
    __builtin_amdgcn_s_wait_tensorcnt((short)0);
  }
  __syncthreads();

  // X[i,k] = sum_t X1[i,t] * C[k,t], C[k,t] = cas(2pi*k*t/nm)*mask[k]*mask[t]
  int kcol = tid & 63;
  int igrp = tid >> 6;               // i in [igrp*32, igrp*32+32)
  float accx[32];
#pragma unroll
  for (int j = 0; j < 32; ++j) accx[j] = 0.f;
  for (int t = 0; t < MM; ++t) {
    if (tid < MM)
      cvec[tid] = casv(twopi * (float)(tid * t) / fnm) * maskf[tid];
    __syncthreads();
    float c = cvec[kcol] * maskf[t];
#pragma unroll
    for (int j = 0; j < 32; ++j)
      accx[j] += Xs[(igrp * 32 + j) * MM + t] * c;
    __syncthreads();
  }
#pragma unroll
  for (int j = 0; j < 32; ++j)
    Xs[(igrp * 32 + j) * MM + kcol] = accx[j];
  __syncthreads();

  int wave = tid >> 5, lane = tid & 31;
  int n = lane & 15, hi = lane >> 4;
  int o0 = wave * 16;
  __builtin_prefetch(Wbig + (size_t)o0 * (MM * CIN), 0, 3);  // L2-hot weights
  v8f zacc[4] = {};
  for (int t = 0; t < MM; ++t) {
    if (tid < MM) {
      int k = tid;
      float mt = maskf[t];
      float c1 = casv(twopi * (float)(k * t) / fnm) * maskf[k] * mt;
      int kf = flip[k];
      float c2 = casv(twopi * (float)(kf * t) / fnm) * maskf[kf] * mt * maskf[k];
      cet[k]  = 0.5f * (c1 + c2);
      cot_[k] = 0.5f * (c1 - c2);
    }
    __syncthreads();
    for (int h = 0; h < 2; ++h) {
      // Bt[(i-h*64)][k] = X[i,k]*Ce[k,t] + Xf[i,k]*Co[k,t]
      for (int j = 0; j < 16; ++j) {
        int idx = j * 256 + tid;
        int ii = idx >> 6, k = idx & 63;
        int i = h * 64 + ii;
        float xv = Xs[i * MM + k];
        float xf = Xs[i * MM + flip[k]] * maskf[k];
        Bt[ii * MM + k] = xv * cet[k] + xf * cot_[k];
      }
      __syncthreads();
      int colbase = t * CIN + h * 64;
      for (int ch = 0; ch < 16; ++ch) {
        int col = colbase + ch * 4 + hi * 2;
        const float* ap = Wbig + (size_t)(o0 + n) * (MM * CIN) + col;
        v2f a; a.x = ap[0]; a.y = ap[1];
#pragma unroll
        for (int kt = 0; kt < 4; ++kt) {
          v2f bf;
          bf.x = Bt[(ch * 4 + hi * 2) * MM + kt * 16 + n];
          bf.y = Bt[(ch * 4 + hi * 2 + 1) * MM + kt * 16 + n];
          zacc[kt] = wmma4(a, bf, zacc[kt]);
        }
      }
      __syncthreads();
    }
  }
#pragma unroll
  for (int kt = 0; kt < 4; ++kt)
#pragma unroll
    for (int v = 0; v < 8; ++v) {
      int o = o0 + v + hi * 8;
      Zm[((size_t)bid * COUT + o) * MM + kt * 16 + n] = zacc[kt][v];
    }
}

// ---------------- K4: res = (Z @ C^T) / nm, WMMA ----------------
__global__ void __launch_bounds__(256)
res_kernel(const float* __restrict__ Zm,
           const int* __restrict__ nmA,
           float* __restrict__ Rm) {
  __shared__ float Cs[MM * MM];     // symmetric C matrix
  int bid = blockIdx.x, tid = threadIdx.x;
  int nm = nmA[bid]; float fnm = (float)nm;
  const float twopi = 6.28318530717958647692f;
  for (int idx = tid; idx < MM * MM; idx += 256) {
    int k = idx >> 6, m = idx & 63;
    float v = 0.f;
    if (k < nm && m < nm) v = casv(twopi * (float)(k * m) / fnm);
    Cs[idx] = v;
  }
  __syncthreads();
  int wave = tid >> 5, lane = tid & 31;
  int n = lane & 15, hi = lane >> 4;
  int o0 = wave * 16;
  v2f areg[16];
#pragma unroll
  for (int ch = 0; ch < 16; ++ch) {
    const float* ap = Zm + ((size_t)bid * COUT + o0 + n) * MM + ch * 4 + hi * 2;
    areg[ch].x = ap[0]; areg[ch].y = ap[1];
  }
  float inv = 1.f / fnm;
  for (int mt = 0; mt < 4; ++mt) {
    v8f acc = {};
#pragma unroll
    for (int ch = 0; ch < 16; ++ch) {
      v2f bf;
      bf.x = Cs[(ch * 4 + hi * 2) * MM + mt * 16 + n];
      bf.y = Cs[(ch * 4 + hi * 2 + 1) * MM + mt * 16 + n];
      acc = wmma4(areg[ch], bf, acc);
    }
#pragma unroll
    for (int v = 0; v < 8; ++v) {
      int o = o0 + v + hi * 8;
      Rm[((size_t)bid * COUT + o) * MM + mt * 16 + n] = acc[v] * inv;
    }
  }
}

// ---------------- K5: inverse DHT (64 modes -> 1024 samples), WMMA ----------------
__global__ void __launch_bounds__(256)
dht_inv_kernel(const float* __restrict__ Rm,
               const float* __restrict__ casML,
               float* __restrict__ out) {
  int bid = blockIdx.x;
  int b = bid >> 3, r = bid & 7;
  int tid = threadIdx.x;
  int wave = tid >> 5, lane = tid & 31;
  int n = lane & 15, hi = lane >> 4;
  int o0 = wave * 16;
  v2f areg[16];
#pragma unroll
  for (int ch = 0; ch < 16; ++ch) {
    const float* ap = Rm + ((size_t)bid * COUT + o0 + n) * MM + ch * 4 + hi * 2;
    areg[ch].x = ap[0]; areg[ch].y = ap[1];
  }
  const float s = 1.f / (float)REG;
  for (int lt = 0; lt < 64; ++lt) {
    int l0 = lt * 16;
    v8f acc = {};
#pragma unroll
    for (int ch = 0; ch < 16; ++ch) {
      v2f bf;
      bf.x = casML[(ch * 4 + hi * 2) * REG + l0 + n];
      bf.y = casML[(ch * 4 + hi * 2 + 1) * REG + l0 + n];
      acc = wmma4(areg[ch], bf, acc);
    }
#pragma unroll
    for (int v = 0; v < 8; ++v) {
      int o = o0 + v + hi * 8;
      out[((size_t)b * COUT + o) * LL + r * REG + l0 + n] = acc[v] * s;
    }
  }
}

extern "C" void kernel_launch(void* const* d_in, const int* in_sizes, int n_in,
                              void* d_out, int out_size, void* d_ws, size_t ws_size,
                              hipStream_t stream) {
  const float* x   = (const float*)d_in[0];   // [16,128,8192]
  const float* err = (const float*)d_in[1];   // [16,8192]
  const float* w   = (const float*)d_in[2];   // [128,128,64]
  float* out = (float*)d_out;                 // [16,128,8192]

  float* casML = (float*)d_ws;                               // 64*1024
  float* casLT = casML + (size_t)MM * REG;                   // 1024*64
  float* Wbig  = casLT + (size_t)MM * REG;                   // 128*8192
  float* X1    = Wbig  + (size_t)COUT * MM * CIN;            // 128*128*64
  float* Zm    = X1    + (size_t)BB * NR * CIN * MM;         // 128*128*64
  float* Rm    = Zm    + (size_t)BB * NR * COUT * MM;        // 128*128*64
  int*   nmA   = (int*)(Rm + (size_t)BB * NR * COUT * MM);   // 128 ints

  prep_kernel<<<1024, 256, 0, stream>>>(w, casML, casLT, Wbig);
  nm_kernel<<<NR, 256, 0, stream>>>(err, nmA);
  dht_fwd_kernel<<<BB * NR, 256, 0, stream>>>(x, casLT, X1);
  zmat_kernel<<<BB * NR, 256, 0, stream>>>(X1, Wbig, nmA, Zm);
  res_kernel<<<BB * NR, 256, 0, stream>>>(Zm, nmA, Rm);
  dht_inv_kernel<<<BB * NR, 256, 0, stream>>>(Rm, casML, out);
}